// MLPBlock_65309272703232
// MI455X (gfx1250) — compile-verified
//
#include <hip/hip_runtime.h>
#include <math.h>

// MoE SwiGLU MLP block for MI455X (gfx1250), fp32 via V_WMMA_F32_16X16X4_F32.
// Sparse (top-4) expert compute with per-expert token compaction:
//   ~29 GFLOP fp32, 113MB of weights (L2-resident on the 192MB L2).
// v2: 32-row M-tiles with K-chunked LDS A staging (halves B VMEM per WMMA),
//     rcp-based sigmoid epilogue.
#define TOK   2048      // B*S
#define HD    768
#define NE    16
#define TOPK  4
#define ID    768
#define I2    1536
#define ALPHA_C 1.702f
#define LIMIT_C 7.0f
#define EPS_C   1e-5f

#define MT    32        // token rows per block
#define KC    256       // K-chunk staged in LDS
#define ASTR  260       // padded row stride: 4*m bank skew, 16B-aligned rows

typedef __attribute__((ext_vector_type(2))) float v2f;
typedef __attribute__((ext_vector_type(8))) float v8f;

__device__ __forceinline__ v8f wmma4(v2f a, v2f b, v8f c) {
  // (neg_a, A, neg_b, B, c_mod, C, reuse_a, reuse_b)
  return __builtin_amdgcn_wmma_f32_16x16x4_f32(false, a, false, b, (short)0, c,
                                               false, false);
}

// ---------------- RMSNorm: x[T,H] -> t_ws[T,H] ----------------
__global__ void rmsnorm_kernel(const float* __restrict__ x,
                               const float* __restrict__ scale,
                               float* __restrict__ t_ws) {
  int t = blockIdx.x;
  const float* xr = x + (size_t)t * HD;
  __shared__ float red[256];
  float s = 0.f;
  for (int c = threadIdx.x; c < HD; c += 256) { float v = xr[c]; s += v * v; }
  red[threadIdx.x] = s;
  __syncthreads();
  for (int o = 128; o > 0; o >>= 1) {
    if (threadIdx.x < o) red[threadIdx.x] += red[threadIdx.x + o];
    __syncthreads();
  }
  float inv = rsqrtf(red[0] / (float)HD + EPS_C);
  float* tr = t_ws + (size_t)t * HD;
  for (int c = threadIdx.x; c < HD; c += 256) tr[c] = xr[c] * inv * scale[c];
}

// ---------------- Router: one wave32 per token ----------------
__global__ void router_kernel(const float* __restrict__ t_ws,
                              const float* __restrict__ gw,
                              const float* __restrict__ gb,
                              float* __restrict__ wts,
                              int* __restrict__ list,
                              int* __restrict__ cnt) {
  int t = blockIdx.x;
  int lane = threadIdx.x;                 // blockDim.x == 32 (one wave)
  const float* tr = t_ws + (size_t)t * HD;
  float acc[NE];
#pragma unroll
  for (int e = 0; e < NE; e++) acc[e] = 0.f;
  for (int h = lane; h < HD; h += 32) {
    float tv = tr[h];
#pragma unroll
    for (int e = 0; e < NE; e++) acc[e] += tv * gw[e * HD + h];
  }
#pragma unroll
  for (int e = 0; e < NE; e++) {
#pragma unroll
    for (int o = 16; o > 0; o >>= 1) acc[e] += __shfl_xor(acc[e], o, 32);
    acc[e] += gb[e];
  }
  // top-4 (ties -> lowest index, matching lax.top_k), redundantly on all lanes
  float vals[TOPK]; int idx[TOPK]; unsigned used = 0u;
#pragma unroll
  for (int k = 0; k < TOPK; k++) {
    float best = -3.4e38f; int bi = 0;
#pragma unroll
    for (int e = 0; e < NE; e++)
      if (!((used >> e) & 1u) && acc[e] > best) { best = acc[e]; bi = e; }
    vals[k] = best; idx[k] = bi; used |= 1u << bi;
  }
  float mx = vals[0], sum = 0.f, w[TOPK];
#pragma unroll
  for (int k = 0; k < TOPK; k++) { w[k] = __expf(vals[k] - mx); sum += w[k]; }
  if (lane == 0) {
    for (int k = 0; k < TOPK; k++) {
      int slot = t * TOPK + k;
      wts[slot] = w[k] / sum;   // softmax; the /K and *K in the ref cancel
      int pos = atomicAdd(&cnt[idx[k]], 1);
      list[idx[k] * TOK + pos] = slot;
    }
  }
}

// ------- MLP1 + fused SwiGLU: gathered [32 x H] @ w1[e] -> act[slot, I] -------
__global__ __launch_bounds__(256)
void mlp1_kernel(const float* __restrict__ t_ws, const float* __restrict__ w1,
                 const float* __restrict__ b1, const int* __restrict__ list,
                 const int* __restrict__ cnt, float* __restrict__ act_ws) {
  int e = blockIdx.z;
  int c = cnt[e];
  int tile0 = blockIdx.y * MT;
  if (tile0 >= c) return;                 // block-uniform early exit

  __shared__ float As[MT][ASTR];          // 32*260*4 = 33280 B
  __shared__ int s_slot[MT];
  if (threadIdx.x < MT) {
    int p = tile0 + threadIdx.x;
    s_slot[threadIdx.x] = list[e * TOK + (p < c ? p : tile0)];
  }

  int lane = threadIdx.x & 31, wave = threadIdx.x >> 5;
  int n = lane & 15, m = lane & 15;
  int koff = (lane >> 4) << 1;            // 0 or 2 (A/B K-pair select)
  int colI = blockIdx.x * 128 + wave * 16;
  const float* Bg = w1 + (size_t)e * HD * I2 + colI + n;  // glu half column
  const float* Bl = Bg + ID;                              // lin half column
  v8f cg0 = {}, cl0 = {}, cg1 = {}, cl1 = {};

  for (int k0 = 0; k0 < HD; k0 += KC) {
    __syncthreads();                      // LDS reuse barrier (also covers s_slot)
    for (int j = threadIdx.x; j < MT * (KC / 4); j += 256) {
      int row = j / (KC / 4), c4 = j % (KC / 4);
      const float4* src =
          (const float4*)(t_ws + (size_t)(s_slot[row] >> 2) * HD + k0);
      *(float4*)&As[row][c4 * 4] = src[c4];
    }
    __syncthreads();
#pragma unroll 4
    for (int kl = 0; kl < KC; kl += 4) {
      int kli = kl + koff;
      int kk = k0 + kli;
      v2f a0 = *(const v2f*)&As[m][kli];
      v2f a1 = *(const v2f*)&As[m + 16][kli];
      v2f bg, bl;
      bg.x = Bg[(size_t)kk * I2];  bg.y = Bg[(size_t)(kk + 1) * I2];
      bl.x = Bl[(size_t)kk * I2];  bl.y = Bl[(size_t)(kk + 1) * I2];
      cg0 = wmma4(a0, bg, cg0);
      cg1 = wmma4(a1, bg, cg1);
      cl0 = wmma4(a0, bl, cl0);
      cl1 = wmma4(a1, bl, cl1);
    }
  }

  float biasg = b1[e * I2 + colI + n];
  float biasl = b1[e * I2 + ID + colI + n];
  int rbase = (lane >> 4) * 8;
#pragma unroll
  for (int r = 0; r < 8; r++) {
    int row0 = r + rbase;                 // rows 0..15
    int row1 = row0 + 16;                 // rows 16..31
    {
      float xg = cg0[r] + biasg;
      float xl = cl0[r] + biasl;
      xg = fminf(xg, LIMIT_C);
      xl = fminf(fmaxf(xl, -LIMIT_C), LIMIT_C);
      float av = xg * __builtin_amdgcn_rcpf(1.f + __expf(-ALPHA_C * xg)) *
                 (xl + 1.f);
      if (tile0 + row0 < c)
        act_ws[(size_t)s_slot[row0] * ID + colI + n] = av;
    }
    {
      float xg = cg1[r] + biasg;
      float xl = cl1[r] + biasl;
      xg = fminf(xg, LIMIT_C);
      xl = fminf(fmaxf(xl, -LIMIT_C), LIMIT_C);
      float av = xg * __builtin_amdgcn_rcpf(1.f + __expf(-ALPHA_C * xg)) *
                 (xl + 1.f);
      if (tile0 + row1 < c)
        act_ws[(size_t)s_slot[row1] * ID + colI + n] = av;
    }
  }
}

// ------- MLP2: gathered act [32 x I] @ w2[e] + b2, scaled by wt -> y[slot, H] -------
__global__ __launch_bounds__(256)
void mlp2_kernel(const float* __restrict__ act_ws, const float* __restrict__ w2,
                 const float* __restrict__ b2, const int* __restrict__ list,
                 const int* __restrict__ cnt, const float* __restrict__ wts,
                 float* __restrict__ y_ws) {
  int e = blockIdx.z;
  int c = cnt[e];
  int tile0 = blockIdx.y * MT;
  if (tile0 >= c) return;

  __shared__ float As[MT][ASTR];
  __shared__ int s_slot[MT];
  __shared__ float s_w[MT];
  if (threadIdx.x < MT) {
    int p = tile0 + threadIdx.x;
    int slot = list[e * TOK + (p < c ? p : tile0)];
    s_slot[threadIdx.x] = slot;
    s_w[threadIdx.x] = wts[slot];
  }

  int lane = threadIdx.x & 31, wave = threadIdx.x >> 5;
  int n = lane & 15, m = lane & 15;
  int koff = (lane >> 4) << 1;
  int colH = blockIdx.x * 128 + wave * 16;
  const float* Bp = w2 + (size_t)e * ID * HD + colH + n;
  v8f c0 = {}, c1 = {};

  for (int k0 = 0; k0 < ID; k0 += KC) {
    __syncthreads();
    for (int j = threadIdx.x; j < MT * (KC / 4); j += 256) {
      int row = j / (KC / 4), c4 = j % (KC / 4);
      const float4* src =
          (const float4*)(act_ws + (size_t)s_slot[row] * ID + k0);
      *(float4*)&As[row][c4 * 4] = src[c4];
    }
    __syncthreads();
#pragma unroll 4
    for (int kl = 0; kl < KC; kl += 4) {
      int kli = kl + koff;
      int kk = k0 + kli;
      v2f a0 = *(const v2f*)&As[m][kli];
      v2f a1 = *(const v2f*)&As[m + 16][kli];
      v2f b;
      b.x = Bp[(size_t)kk * HD];  b.y = Bp[(size_t)(kk + 1) * HD];
      c0 = wmma4(a0, b, c0);
      c1 = wmma4(a1, b, c1);
    }
  }

  float bias = b2[e * HD + colH + n];
  int rbase = (lane >> 4) * 8;
#pragma unroll
  for (int r = 0; r < 8; r++) {
    int row0 = r + rbase;
    int row1 = row0 + 16;
    float y0 = (c0[r] + bias) * s_w[row0];
    float y1 = (c1[r] + bias) * s_w[row1];
    if (tile0 + row0 < c) y_ws[(size_t)s_slot[row0] * HD + colH + n] = y0;
    if (tile0 + row1 < c) y_ws[(size_t)s_slot[row1] * HD + colH + n] = y1;
  }
}

// ---------------- Combine: out = x + sum_k y[t*4+k]  (fixed order) ----------------
__global__ void combine_kernel(const float* __restrict__ x,
                               const float* __restrict__ y_ws,
                               float* __restrict__ out) {
  int t = blockIdx.x;
  for (int h = threadIdx.x; h < HD; h += 256) {
    float acc = x[(size_t)t * HD + h];
#pragma unroll
    for (int k = 0; k < TOPK; k++) acc += y_ws[(size_t)(t * TOPK + k) * HD + h];
    out[(size_t)t * HD + h] = acc;
  }
}

extern "C" void kernel_launch(void* const* d_in, const int* in_sizes, int n_in,
                              void* d_out, int out_size, void* d_ws, size_t ws_size,
                              hipStream_t stream) {
  const float* x     = (const float*)d_in[0];
  const float* scale = (const float*)d_in[1];
  const float* gw    = (const float*)d_in[2];
  const float* gb    = (const float*)d_in[3];
  const float* w1    = (const float*)d_in[4];
  const float* b1    = (const float*)d_in[5];
  const float* w2    = (const float*)d_in[6];
  const float* b2    = (const float*)d_in[7];
  float* out = (float*)d_out;

  // Workspace layout (~57 MB total)
  char* p = (char*)d_ws;
  float* t_ws   = (float*)p;  p += (size_t)TOK * HD * 4;          // 6.29 MB
  float* wts    = (float*)p;  p += (size_t)TOK * TOPK * 4;        // 32 KB
  int*   list   = (int*)p;    p += (size_t)NE * TOK * 4;          // 128 KB
  int*   cnt    = (int*)p;    p += 256;                           // aligned pad
  float* act_ws = (float*)p;  p += (size_t)TOK * TOPK * ID * 4;   // 25.2 MB
  float* y_ws   = (float*)p;                                      // 25.2 MB

  hipMemsetAsync(cnt, 0, NE * sizeof(int), stream);
  rmsnorm_kernel<<<TOK, 256, 0, stream>>>(x, scale, t_ws);
  router_kernel<<<TOK, 32, 0, stream>>>(t_ws, gw, gb, wts, list, cnt);
  mlp1_kernel<<<dim3(ID / 128, TOK / MT, NE), 256, 0, stream>>>(
      t_ws, w1, b1, list, cnt, act_ws);
  mlp2_kernel<<<dim3(HD / 128, TOK / MT, NE), 256, 0, stream>>>(
      act_ws, w2, b2, list, cnt, wts, y_ws);
  combine_kernel<<<TOK, 256, 0, stream>>>(x, y_ws, out);
}